// CNN_73306501808283
// MI455X (gfx1250) — compile-verified
//
#include <hip/hip_runtime.h>
#include <hip/hip_bf16.h>

typedef __attribute__((ext_vector_type(2))) float v2f;
typedef __attribute__((ext_vector_type(8))) float v8f;

#define BN_EPS 1e-5f

// ---------------- Workspace layout (in floats) ----------------
// ws0: conv1 out [64,5,488,8]  = 1,249,280
// ws1: pool1 out [64,5,122,2]  =    78,080
// ws2: conv2 out [64,10,118,2] =   151,040
// ws3: pool2 out [64,10,59,1]  =    37,760  (== flattened [64,590])
// ws4: fc pre    [64,10]       =       640
// st1: 5 channels * {mean,var} =        10
// st2: 10 channels * {mean,var}=        20
#define OFF0 0
#define OFF1 1249280
#define OFF2 1327360
#define OFF3 1478400
#define OFF4 1516160
#define OFFS1 1516800
#define OFFS2 1516810

// ---------------------------------------------------------------
// Kernel 1: conv1 (AdderNet L1 conv), x:[64,1,499,12] w1:[5,1,12,5]
// out:[64,5,488,8].  grid=(16, 64*5), block=256, one thread/output.
// ---------------------------------------------------------------
__global__ __launch_bounds__(256) void conv1_kernel(
    const float* __restrict__ x, const float* __restrict__ w1,
    float* __restrict__ out) {
  __shared__ float ws[60];
  const int bc = blockIdx.y;          // b*5 + c
  const int b  = bc / 5, c = bc % 5;
  if (threadIdx.x < 60) ws[threadIdx.x] = w1[c * 60 + threadIdx.x];
  __syncthreads();
  int sp = blockIdx.x * 256 + threadIdx.x;   // 0..3903 spatial
  if (sp >= 488 * 8) return;
  int y  = sp >> 3;                   // 0..487
  int xo = sp & 7;                    // 0..7
  const float* xp = x + (b * 499 + y) * 12 + xo;
  __builtin_prefetch(xp + 16 * 12, 0, 3);   // gfx1250 global_prefetch_b8
  float acc = 0.f;
#pragma unroll
  for (int i = 0; i < 12; ++i) {
#pragma unroll
    for (int j = 0; j < 5; ++j) {
      acc += fabsf(xp[i * 12 + j] - ws[i * 5 + j]);
    }
  }
  out[bc * (488 * 8) + sp] = -acc;
}

// ---------------------------------------------------------------
// Per-channel mean / biased-var over [B, C, S] data. One block per
// channel, fixed-order strided accumulation + LDS tree: deterministic.
// ---------------------------------------------------------------
__global__ __launch_bounds__(256) void channel_stats(
    const float* __restrict__ data, float* __restrict__ stats,
    int per_channel, int nchan, int batch) {
  const int c = blockIdx.x;
  float s = 0.f, s2 = 0.f;
  const int total = batch * per_channel;
  for (int i = threadIdx.x; i < total; i += 256) {
    int b = i / per_channel, sp = i - b * per_channel;
    float v = data[(b * nchan + c) * per_channel + sp];
    s += v; s2 += v * v;
  }
  __shared__ float sh[256], sh2[256];
  sh[threadIdx.x] = s; sh2[threadIdx.x] = s2;
  __syncthreads();
  for (int o = 128; o > 0; o >>= 1) {
    if (threadIdx.x < o) {
      sh[threadIdx.x]  += sh[threadIdx.x + o];
      sh2[threadIdx.x] += sh2[threadIdx.x + o];
    }
    __syncthreads();
  }
  if (threadIdx.x == 0) {
    float n = (float)total;
    float mean = sh[0] / n;
    float m2   = sh2[0] / n;
    stats[c * 2 + 0] = mean;
    stats[c * 2 + 1] = m2 - mean * mean;   // biased variance
  }
}

// ---------------------------------------------------------------
// Kernel: BN + ReLU + maxpool 4x4.  in:[64,5,488,8] -> out:[64,5,122,2]
// ---------------------------------------------------------------
__global__ __launch_bounds__(256) void bn_relu_pool1(
    const float* __restrict__ in, const float* __restrict__ stats,
    const float* __restrict__ g, const float* __restrict__ bt,
    float* __restrict__ out) {
  int idx = blockIdx.x * 256 + threadIdx.x;
  if (idx >= 64 * 5 * 122 * 2) return;
  int px = idx & 1;
  int py = (idx >> 1) % 122;
  int bc = idx / 244;
  int c  = bc % 5;
  float mean = stats[c * 2], var = stats[c * 2 + 1];
  float inv = rsqrtf(var + BN_EPS);
  float sc = g[c] * inv;
  float sb = bt[c] - mean * sc;
  const float* p = in + (bc * 488 + py * 4) * 8 + px * 4;
  float m = 0.f;   // relu(..) >= 0, so 0 is a safe identity for the max
#pragma unroll
  for (int dy = 0; dy < 4; ++dy)
#pragma unroll
    for (int dx = 0; dx < 4; ++dx) {
      float v = fmaxf(p[dy * 8 + dx] * sc + sb, 0.f);
      m = fmaxf(m, v);
    }
  out[idx] = m;
}

// ---------------------------------------------------------------
// Kernel: conv2.  in:[64,5,122,2] w2:[10,5,5,1] -> out:[64,10,118,2]
// ---------------------------------------------------------------
__global__ __launch_bounds__(256) void conv2_kernel(
    const float* __restrict__ in, const float* __restrict__ w2,
    float* __restrict__ out) {
  __shared__ float ws[250];
  if (threadIdx.x < 250) ws[threadIdx.x] = w2[threadIdx.x];
  __syncthreads();
  int idx = blockIdx.x * 256 + threadIdx.x;
  if (idx >= 64 * 10 * 118 * 2) return;
  int x  = idx & 1;
  int y  = (idx >> 1) % 118;
  int bc = idx / 236;
  int c  = bc % 10, b = bc / 10;
  float acc = 0.f;
#pragma unroll
  for (int cin = 0; cin < 5; ++cin) {
    const float* p = in + ((b * 5 + cin) * 122 + y) * 2 + x;
#pragma unroll
    for (int i = 0; i < 5; ++i)
      acc += fabsf(p[i * 2] - ws[(c * 5 + cin) * 5 + i]);
  }
  out[idx] = -acc;
}

// ---------------------------------------------------------------
// Kernel: BN + ReLU + maxpool 2x2.  in:[64,10,118,2] -> out:[64,10,59,1]
// ---------------------------------------------------------------
__global__ __launch_bounds__(256) void bn_relu_pool2(
    const float* __restrict__ in, const float* __restrict__ stats,
    const float* __restrict__ g, const float* __restrict__ bt,
    float* __restrict__ out) {
  int idx = blockIdx.x * 256 + threadIdx.x;
  if (idx >= 64 * 10 * 59) return;
  int py = idx % 59;
  int bc = idx / 59;
  int c  = bc % 10;
  float mean = stats[c * 2], var = stats[c * 2 + 1];
  float inv = rsqrtf(var + BN_EPS);
  float sc = g[c] * inv;
  float sb = bt[c] - mean * sc;
  const float* p = in + (bc * 118 + py * 2) * 2;  // 2x2 window, contiguous 4
  float m = 0.f;
#pragma unroll
  for (int d = 0; d < 4; ++d)
    m = fmaxf(m, fmaxf(p[d] * sc + sb, 0.f));
  out[idx] = m;
}

// ---------------------------------------------------------------
// Kernel: FC adder layer via WMMA f32 16x16x4.
// pre[b,c] = -sum_k |x[b,k] - w[c,k]|, K=590, x:[64,590] w:[10,590].
// 4 blocks (16-batch tiles) x 10 waves (one wave per output channel).
// A = |x - w| tile (16 rows x 4 k), B = ones => each D column holds
// the row sums; accumulate over 148 k-chunks in the C operand.
// A layout (32b 16x4): lanes 0-15 hold K={kk,kk+1}, lanes 16-31 K={kk+2,kk+3},
// row M = lane%16.  D: lane 0 VGPR r -> row r col 0; lane 16 -> row 8+r.
// ---------------------------------------------------------------
__global__ __launch_bounds__(320) void fc_wmma_kernel(
    const float* __restrict__ xin,   // [64,590]
    const float* __restrict__ wfc,   // [10,590]
    float* __restrict__ pre) {       // [64,10]
  const int lane = threadIdx.x & 31;
  const int c    = threadIdx.x >> 5;     // wave id == channel 0..9
  const int m    = lane & 15;            // row within 16-batch tile
  const int hi   = lane >> 4;            // k sub-group
  const int b    = blockIdx.x * 16 + m;
  const float* xrow = xin + b * 590;
  const float* wrow = wfc + c * 590;
  v8f acc = {0.f, 0.f, 0.f, 0.f, 0.f, 0.f, 0.f, 0.f};
  v2f ones; ones.x = 1.0f; ones.y = 1.0f;
  for (int kk = 0; kk < 590; kk += 4) {
    int k0 = kk + 2 * hi;
    int k1 = k0 + 1;
    int kc0 = (k0 < 590) ? k0 : 589;     // clamp; mask result instead of branching
    int kc1 = (k1 < 590) ? k1 : 589;
    float m0 = (k0 < 590) ? 1.f : 0.f;
    float m1 = (k1 < 590) ? 1.f : 0.f;
    v2f a;
    a.x = m0 * fabsf(xrow[kc0] - wrow[kc0]);
    a.y = m1 * fabsf(xrow[kc1] - wrow[kc1]);
    acc = __builtin_amdgcn_wmma_f32_16x16x4_f32(
        /*neg_a=*/false, a, /*neg_b=*/false, ones,
        /*c_mod=*/(short)0, acc, /*reuse_a=*/false, /*reuse_b=*/false);
  }
  if (lane == 0 || lane == 16) {
    int rbase = blockIdx.x * 16 + hi * 8;
#pragma unroll
    for (int r = 0; r < 8; ++r)
      pre[(rbase + r) * 10 + c] = -acc[r];
  }
}

// ---------------------------------------------------------------
// Kernel: final BN over batch dim only. pre:[64,10] -> out:[64,10]
// ---------------------------------------------------------------
__global__ __launch_bounds__(256) void fc_bn_final(
    const float* __restrict__ pre, const float* __restrict__ g,
    const float* __restrict__ bt, float* __restrict__ out) {
  __shared__ float sc_s[10], sb_s[10];
  int t = threadIdx.x;
  if (t < 10) {
    float s = 0.f, s2 = 0.f;
    for (int b = 0; b < 64; ++b) {
      float v = pre[b * 10 + t];
      s += v; s2 += v * v;
    }
    float mean = s / 64.f;
    float var  = s2 / 64.f - mean * mean;
    float inv  = rsqrtf(var + BN_EPS);
    float sc   = g[t] * inv;
    sc_s[t] = sc;
    sb_s[t] = bt[t] - mean * sc;
  }
  __syncthreads();
  for (int i = t; i < 640; i += 256) {
    int c = i % 10;
    out[i] = pre[i] * sc_s[c] + sb_s[c];
  }
}

// ---------------------------------------------------------------
extern "C" void kernel_launch(void* const* d_in, const int* in_sizes, int n_in,
                              void* d_out, int out_size, void* d_ws, size_t ws_size,
                              hipStream_t stream) {
  const float* x    = (const float*)d_in[0];  // [64,1,499,12]
  const float* w1   = (const float*)d_in[1];  // [5,1,12,5]
  const float* g1   = (const float*)d_in[2];
  const float* b1   = (const float*)d_in[3];
  const float* w2   = (const float*)d_in[4];  // [10,5,5,1]
  const float* g2   = (const float*)d_in[5];
  const float* b2   = (const float*)d_in[6];
  const float* w_fc = (const float*)d_in[7];  // [10,590]
  const float* g3   = (const float*)d_in[8];
  const float* b3   = (const float*)d_in[9];
  float* out = (float*)d_out;

  float* ws  = (float*)d_ws;
  float* ws0 = ws + OFF0;   // conv1 out
  float* ws1 = ws + OFF1;   // pool1 out
  float* ws2 = ws + OFF2;   // conv2 out
  float* ws3 = ws + OFF3;   // pool2 out (== [64,590])
  float* ws4 = ws + OFF4;   // fc pre
  float* st1 = ws + OFFS1;
  float* st2 = ws + OFFS2;

  // 1) conv1: [64,5,488,8]
  conv1_kernel<<<dim3(16, 64 * 5), 256, 0, stream>>>(x, w1, ws0);
  // 2) stats over (B,H,W) for 5 channels, S=3904
  channel_stats<<<5, 256, 0, stream>>>(ws0, st1, 488 * 8, 5, 64);
  // 3) BN + ReLU + pool4 -> [64,5,122,2]
  bn_relu_pool1<<<(64 * 5 * 122 * 2 + 255) / 256, 256, 0, stream>>>(ws0, st1, g1, b1, ws1);
  // 4) conv2 -> [64,10,118,2]
  conv2_kernel<<<(64 * 10 * 118 * 2 + 255) / 256, 256, 0, stream>>>(ws1, w2, ws2);
  // 5) stats for 10 channels, S=236
  channel_stats<<<10, 256, 0, stream>>>(ws2, st2, 118 * 2, 10, 64);
  // 6) BN + ReLU + pool2 -> [64,10,59,1]
  bn_relu_pool2<<<(64 * 10 * 59 + 255) / 256, 256, 0, stream>>>(ws2, st2, g2, b2, ws3);
  // 7) FC adder layer via WMMA f32 -> pre[64,10]
  fc_wmma_kernel<<<4, 320, 0, stream>>>(ws3, w_fc, ws4);
  // 8) final batch-only BN -> out[64,10]
  fc_bn_final<<<1, 256, 0, stream>>>(ws4, g3, b3, out);
}